// GIN_65094524338980
// MI455X (gfx1250) — compile-verified
//
#include <hip/hip_runtime.h>
#include <hip/hip_bf16.h>
#include <math.h>

// ---------------------------------------------------------------------------
// GIN node classification on gfx1250 (MI455X).
// fp32 end-to-end; dense layers use V_WMMA_F32_16X16X4_F32 (wave32 WMMA).
// Block = 4 waves = one 16-row m-tile; A-tile and weights staged in LDS
// (weights via GLOBAL_LOAD_ASYNC_TO_LDS_B128 when available).
// Edge scatter stays L2-resident (192MB L2 >> 25.6MB agg buffer).
// ---------------------------------------------------------------------------

#define N_NODES 50000
#define N_EDGES 800000
#define N_FEAT  128
#define N_HID   64
#define N_EMB   128
#define N_CLASS 16
#define BN_EPS  1e-5f

#define ASTRIDE 132  // 16-row A tile row pad (132%64==4 -> conflict-free frags)
#define WSTRIDE 68   // weight row pad (68*4 bytes %16==0 -> aligned b128)

typedef __attribute__((ext_vector_type(2))) float v2f;
typedef __attribute__((ext_vector_type(8))) float v8f;
typedef __attribute__((ext_vector_type(4))) int v4i;
// async-copy operand types: global-src / LDS-dst pointers to 128-bit payloads
typedef __attribute__((address_space(1))) v4i g_v4i;
typedef __attribute__((address_space(3))) v4i l_v4i;

#if defined(__has_builtin)
#if __has_builtin(__builtin_amdgcn_global_load_async_to_lds_b128)
#define HAVE_ASYNC_LDS 1
#endif
#endif

__device__ __forceinline__ v8f wmma_f32_k4(v2f a, v2f b, v8f c) {
  // 8 args: (neg_a, A, neg_b, B, c_mod, C, reuse_a, reuse_b)
  return __builtin_amdgcn_wmma_f32_16x16x4_f32(false, a, false, b, (short)0, c,
                                               false, false);
}

__device__ __forceinline__ void wait_asynccnt0() {
#if defined(HAVE_ASYNC_LDS)
#if __has_builtin(__builtin_amdgcn_s_wait_asynccnt)
  __builtin_amdgcn_s_wait_asynccnt(0);
#else
  asm volatile("s_wait_asynccnt 0x0" ::: "memory");
#endif
#endif
}

// Stage W[K,64] (row-major) into LDS rows padded to WSTRIDE.
__device__ __forceinline__ void stage_weights(const float* __restrict__ W,
                                              float* __restrict__ Wl, int K,
                                              int t) {
#if defined(HAVE_ASYNC_LDS)
  for (int idx = t; idx < K * 16; idx += 128) {  // 16 float4 chunks per row
    const int k = idx >> 4;
    const int n4 = (idx & 15) << 2;
    __builtin_amdgcn_global_load_async_to_lds_b128(
        (g_v4i*)(W + (size_t)k * 64 + n4),
        (l_v4i*)(Wl + k * WSTRIDE + n4), 0, 0);
  }
  wait_asynccnt0();
#else
  for (int idx = t; idx < K * 16; idx += 128) {
    const int k = idx >> 4;
    const int n4 = (idx & 15) << 2;
    *(float4*)(Wl + k * WSTRIDE + n4) =
        *(const float4*)(W + (size_t)k * 64 + n4);
  }
#endif
}

// --------------------------- utility: zero fill ----------------------------
__global__ void zero_f32(float* __restrict__ p, int n) {
  int i = blockIdx.x * blockDim.x + threadIdx.x;
  if (i < n) p[i] = 0.0f;
}

// ------------------------ edge scatter (segment_sum) -----------------------
// One thread handles one (edge, 4-channel chunk): float4 gather + 4 atomics.
__global__ void scatter_add(const float* __restrict__ Hsrc,
                            const int* __restrict__ src,
                            const int* __restrict__ dst,
                            float* __restrict__ AGG, int dim) {
  const int chunks = dim >> 2;
  const long long tid = (long long)blockIdx.x * blockDim.x + threadIdx.x;
  const long long total = (long long)N_EDGES * chunks;
  if (tid >= total) return;
  const int e  = (int)(tid / chunks);
  const int c0 = (int)(tid % chunks) << 2;
  const int s = src[e];
  const int d = dst[e];
  const float4 v = *(const float4*)(Hsrc + (size_t)s * dim + c0);
  float* out = AGG + (size_t)d * dim + c0;
  atomicAdd(out + 0, v.x);
  atomicAdd(out + 1, v.y);
  atomicAdd(out + 2, v.z);
  atomicAdd(out + 3, v.w);
}

// ------------------- GEMM1: Z = (H + AGG) @ W[K,64] + b --------------------
// One block (4 waves) per 16-row m-tile; wave w computes n-tile w.
// A = (H+AGG) staged in LDS via coalesced float4 loads; W async-staged in LDS.
// f32 16x16x4 frag layout: lanes 0-15: M=lane, K=k0+{0,1}; lanes 16-31: K=k0+{2,3}.
// C/D: VGPR r -> (M = half*8 + r, N = lane&15).
__global__ void gin_mlp1(const float* __restrict__ H,
                         const float* __restrict__ AGG,
                         const float* __restrict__ W,
                         const float* __restrict__ bias,
                         float* __restrict__ Z, int K) {
  __shared__ float As[16 * ASTRIDE];
  __shared__ float Wl[128 * WSTRIDE];
  const int t = threadIdx.x;
  const int m0 = blockIdx.x << 4;

  stage_weights(W, Wl, K, t);

  const int cpr = K >> 2;  // float4 chunks per A row
  for (int idx = t; idx < 16 * cpr; idx += 128) {
    const int r = idx / cpr;
    const int c4 = (idx % cpr) << 2;
    const float4 hv = *(const float4*)(H + (size_t)(m0 + r) * K + c4);
    const float4 av = *(const float4*)(AGG + (size_t)(m0 + r) * K + c4);
    float4 s;
    s.x = hv.x + av.x; s.y = hv.y + av.y;
    s.z = hv.z + av.z; s.w = hv.w + av.w;
    *(float4*)(As + r * ASTRIDE + c4) = s;
  }
  __syncthreads();

  const int lane = t & 31;
  const int n0 = (t >> 5) << 4;  // wave -> n-tile
  const int half = lane >> 4;
  const int l = lane & 15;
  v8f acc = {};
  for (int k0 = 0; k0 < K; k0 += 4) {
    const int ka = k0 + 2 * half;
    v2f a, b;
    a.x = As[l * ASTRIDE + ka];
    a.y = As[l * ASTRIDE + ka + 1];
    b.x = Wl[ka * WSTRIDE + n0 + l];
    b.y = Wl[(ka + 1) * WSTRIDE + n0 + l];
    acc = wmma_f32_k4(a, b, acc);
  }
  const float bv = bias[n0 + l];
  float* zp = Z + (size_t)(m0 + half * 8) * N_HID + n0 + l;
#pragma unroll
  for (int r = 0; r < 8; ++r) zp[(size_t)r * N_HID] = acc[r] + bv;
}

// ----------------- BN statistics: per-channel sum / sumsq ------------------
__global__ void bn_stats(const float* __restrict__ Z, float* __restrict__ stats,
                         int M) {
  __shared__ float ssum[4][64];
  __shared__ float ssq[4][64];
  const int c = threadIdx.x & 63;
  const int g = threadIdx.x >> 6;  // 0..3
  float s = 0.0f, s2 = 0.0f;
  for (int r = blockIdx.x * 4 + g; r < M; r += gridDim.x * 4) {
    const float v = Z[(size_t)r * N_HID + c];
    s += v;
    s2 += v * v;
  }
  ssum[g][c] = s;
  ssq[g][c] = s2;
  __syncthreads();
  if (threadIdx.x < 64) {
    float ts = 0.0f, tq = 0.0f;
#pragma unroll
    for (int i = 0; i < 4; ++i) {
      ts += ssum[i][threadIdx.x];
      tq += ssq[i][threadIdx.x];
    }
    atomicAdd(&stats[threadIdx.x], ts);
    atomicAdd(&stats[64 + threadIdx.x], tq);
  }
}

// ---------- GEMM2: Hout = relu( relu(BN(Z)) @ W2[64,64] + b2 ) -------------
// BN scale/shift precomputed once per block; applied while staging A in LDS.
__global__ void gin_mlp2(const float* __restrict__ Z,
                         const float* __restrict__ stats,
                         const float* __restrict__ gamma,
                         const float* __restrict__ beta,
                         const float* __restrict__ W2,
                         const float* __restrict__ b2,
                         float* __restrict__ Hout, int M) {
  __shared__ float As[16 * ASTRIDE];
  __shared__ float Wl[64 * WSTRIDE];
  __shared__ float sc[64];
  __shared__ float sh[64];
  const int t = threadIdx.x;
  const int m0 = blockIdx.x << 4;

  stage_weights(W2, Wl, N_HID, t);

  if (t < 64) {  // BN affine (training-mode batch stats, biased variance)
    const float invM = 1.0f / (float)M;
    const float mu = stats[t] * invM;
    const float va = stats[64 + t] * invM - mu * mu;
    const float s = gamma[t] * rsqrtf(va + BN_EPS);
    sc[t] = s;
    sh[t] = beta[t] - mu * s;
  }
  __syncthreads();

  for (int idx = t; idx < 16 * 16; idx += 128) {  // 16 rows x 16 float4 chunks
    const int r = idx >> 4;
    const int c4 = (idx & 15) << 2;
    const float4 zv = *(const float4*)(Z + (size_t)(m0 + r) * N_HID + c4);
    float4 s;
    s.x = fmaxf(fmaf(zv.x, sc[c4 + 0], sh[c4 + 0]), 0.0f);
    s.y = fmaxf(fmaf(zv.y, sc[c4 + 1], sh[c4 + 1]), 0.0f);
    s.z = fmaxf(fmaf(zv.z, sc[c4 + 2], sh[c4 + 2]), 0.0f);
    s.w = fmaxf(fmaf(zv.w, sc[c4 + 3], sh[c4 + 3]), 0.0f);
    *(float4*)(As + r * ASTRIDE + c4) = s;
  }
  __syncthreads();

  const int lane = t & 31;
  const int n0 = (t >> 5) << 4;
  const int half = lane >> 4;
  const int l = lane & 15;
  v8f acc = {};
  for (int k0 = 0; k0 < N_HID; k0 += 4) {
    const int ka = k0 + 2 * half;
    v2f a, b;
    a.x = As[l * ASTRIDE + ka];
    a.y = As[l * ASTRIDE + ka + 1];
    b.x = Wl[ka * WSTRIDE + n0 + l];
    b.y = Wl[(ka + 1) * WSTRIDE + n0 + l];
    acc = wmma_f32_k4(a, b, acc);
  }
  const float bv = b2[n0 + l];
  float* hp = Hout + (size_t)(m0 + half * 8) * N_HID + n0 + l;
#pragma unroll
  for (int r = 0; r < 8; ++r) hp[(size_t)r * N_HID] = fmaxf(acc[r] + bv, 0.0f);
}

// ------- head: out = sigmoid( relu(H@Wl1 + bl1) @ Wl2 + bl2 ) --------------
// Embedding tile (16x128) staged in padded LDS (stride 130 avoids bank
// conflicts); second GEMM reads A-frags from LDS. Per-wave slab, explicit
// s_wait_dscnt between the stages.
__global__ void head_fused(const float* __restrict__ H,
                           const float* __restrict__ Wl1,
                           const float* __restrict__ bl1,
                           const float* __restrict__ Wl2,
                           const float* __restrict__ bl2,
                           float* __restrict__ out, int M) {
  __shared__ float lds[4 * 16 * 130];
  const int lane = threadIdx.x & 31;
  const int wave = threadIdx.x >> 5;
  const int mt = blockIdx.x * 4 + wave;
  if (mt >= (M >> 4)) return;  // wave-uniform; no __syncthreads used here
  const int m0 = mt << 4;
  const int half = lane >> 4;
  const int l = lane & 15;
  float* slab = lds + wave * (16 * 130);

  // stage 1: E[16,128] = relu(H[16,64] @ Wl1[64,128] + bl1) -> LDS
  for (int nt = 0; nt < (N_EMB >> 4); ++nt) {
    const int n0 = nt << 4;
    v8f acc = {};
    for (int k0 = 0; k0 < N_HID; k0 += 4) {
      const int ka = k0 + 2 * half;
      v2f a, b;
      a.x = H[(size_t)(m0 + l) * N_HID + ka];
      a.y = H[(size_t)(m0 + l) * N_HID + ka + 1];
      b.x = Wl1[(size_t)ka * N_EMB + n0 + l];
      b.y = Wl1[(size_t)(ka + 1) * N_EMB + n0 + l];
      acc = wmma_f32_k4(a, b, acc);
    }
    const float bv = bl1[n0 + l];
#pragma unroll
    for (int r = 0; r < 8; ++r)
      slab[(half * 8 + r) * 130 + n0 + l] = fmaxf(acc[r] + bv, 0.0f);
  }
  asm volatile("s_wait_dscnt 0" ::: "memory");

  // stage 2: out[16,16] = sigmoid(E[16,128] @ Wl2[128,16] + bl2)
  v8f acc = {};
  for (int k0 = 0; k0 < N_EMB; k0 += 4) {
    const int ka = k0 + 2 * half;
    v2f a, b;
    a.x = slab[l * 130 + ka];
    a.y = slab[l * 130 + ka + 1];
    b.x = Wl2[(size_t)ka * N_CLASS + l];
    b.y = Wl2[(size_t)(ka + 1) * N_CLASS + l];
    acc = wmma_f32_k4(a, b, acc);
  }
  const float bv = bl2[l];
#pragma unroll
  for (int r = 0; r < 8; ++r) {
    const float v = acc[r] + bv;
    out[(size_t)(m0 + half * 8 + r) * N_CLASS + l] = 1.0f / (1.0f + __expf(-v));
  }
}

// ---------------------------------------------------------------------------
extern "C" void kernel_launch(void* const* d_in, const int* in_sizes, int n_in,
                              void* d_out, int out_size, void* d_ws,
                              size_t ws_size, hipStream_t stream) {
  const float* x   = (const float*)d_in[0];
  const int*   ei  = (const int*)d_in[1];  // [2, N_EDGES]: row0=src, row1=dst
  const float* W1a = (const float*)d_in[2];
  const float* b1a = (const float*)d_in[3];
  const float* ga  = (const float*)d_in[4];
  const float* ba  = (const float*)d_in[5];
  const float* W2a = (const float*)d_in[6];
  const float* b2a = (const float*)d_in[7];
  const float* Ws1 = (const float*)d_in[8];   // [2,64,64]
  const float* bs1 = (const float*)d_in[9];   // [2,64]
  const float* gs  = (const float*)d_in[10];  // [2,64]
  const float* bes = (const float*)d_in[11];  // [2,64]
  const float* Ws2 = (const float*)d_in[12];  // [2,64,64]
  const float* bs2 = (const float*)d_in[13];  // [2,64]
  const float* Wl1 = (const float*)d_in[14];  // [64,128]
  const float* bl1 = (const float*)d_in[15];
  const float* Wl2 = (const float*)d_in[16];  // [128,16]
  const float* bl2 = (const float*)d_in[17];
  float* outp = (float*)d_out;

  const int* srcI = ei;
  const int* dstI = ei + N_EDGES;

  // workspace carve-up (all fully rewritten each call: deterministic)
  float* h     = (float*)d_ws;                 // [N_NODES,64]
  float* z     = h + (size_t)N_NODES * N_HID;  // [N_NODES,64]
  float* agg   = z + (size_t)N_NODES * N_HID;  // [N_NODES,128] (reused 64-wide)
  float* stats = agg + (size_t)N_NODES * N_FEAT;  // [128] sum|sumsq

  const int mtiles = N_NODES >> 4;  // 3125 blocks, 4 waves each

  // ---------------- conv1: dim 128 -> 64 ----------------
  {
    const int n = N_NODES * N_FEAT;
    zero_f32<<<(n + 255) / 256, 256, 0, stream>>>(agg, n);
    zero_f32<<<1, 128, 0, stream>>>(stats, 128);
    const long long sthreads = (long long)N_EDGES * (N_FEAT >> 2);
    scatter_add<<<(int)((sthreads + 255) / 256), 256, 0, stream>>>(
        x, srcI, dstI, agg, N_FEAT);
    gin_mlp1<<<mtiles, 128, 0, stream>>>(x, agg, W1a, b1a, z, N_FEAT);
    bn_stats<<<256, 256, 0, stream>>>(z, stats, N_NODES);
    gin_mlp2<<<mtiles, 128, 0, stream>>>(z, stats, ga, ba, W2a, b2a, h,
                                         N_NODES);
  }

  // ---------------- extra GIN layers: dim 64 -> 64 ----------------
  for (int i = 0; i < 2; ++i) {
    const int n = N_NODES * N_HID;
    zero_f32<<<(n + 255) / 256, 256, 0, stream>>>(agg, n);
    zero_f32<<<1, 128, 0, stream>>>(stats, 128);
    const long long sthreads = (long long)N_EDGES * (N_HID >> 2);
    scatter_add<<<(int)((sthreads + 255) / 256), 256, 0, stream>>>(
        h, srcI, dstI, agg, N_HID);
    gin_mlp1<<<mtiles, 128, 0, stream>>>(h, agg, Ws1 + (size_t)i * 64 * 64,
                                         bs1 + (size_t)i * 64, z, N_HID);
    bn_stats<<<256, 256, 0, stream>>>(z, stats, N_NODES);
    gin_mlp2<<<mtiles, 128, 0, stream>>>(
        z, stats, gs + (size_t)i * 64, bes + (size_t)i * 64,
        Ws2 + (size_t)i * 64 * 64, bs2 + (size_t)i * 64, h, N_NODES);
  }

  // ---------------- head: 64 -> 128 -> 16, sigmoid ----------------
  head_fused<<<(mtiles + 3) / 4, 128, 0, stream>>>(h, Wl1, bl1, Wl2, bl2, outp,
                                                   N_NODES);
}